// ExemplarSoftmaxLossORG_17102559773291
// MI455X (gfx1250) — compile-verified
//
#include <hip/hip_runtime.h>
#include <hip/hip_bf16.h>
#include <math.h>

// ---------------------------------------------------------------------------
// Problem constants (from reference: B=8192, D=256, C=1000)
// ---------------------------------------------------------------------------
#define B_ROWS 8192
#define DIMS   256
#define NCLS   1000
#define ROWS3  (3 * B_ROWS)
#define EPS_PD 1e-6f
#define CENTER_W 0.001f

#define CE_BLOCKS 512
#define CE_WPB    8                                   // waves per block
#define CE_ROWS_PER_WAVE (ROWS3 / (CE_BLOCKS * CE_WPB)) // 24576/4096 = 6
#define F4_PER_ROW 250                                // 1000 floats / 4

typedef float v2f __attribute__((ext_vector_type(2)));
typedef float v8f __attribute__((ext_vector_type(8)));
typedef int   v4i __attribute__((vector_size(4 * sizeof(int))));  // matches builtin proto

// ---------------------------------------------------------------------------
// CDNA5 feature probes (compile-clean either way)
// ---------------------------------------------------------------------------
#if defined(__gfx1250__) && __has_builtin(__builtin_amdgcn_wmma_f32_16x16x4_f32)
#define HAS_WMMA_F32X4 1
#else
#define HAS_WMMA_F32X4 0
#endif

#if defined(__gfx1250__) && __has_builtin(__builtin_amdgcn_global_load_async_to_lds_b128)
#define HAS_ASYNC_LDS 1
#else
#define HAS_ASYNC_LDS 0
#endif

#if HAS_ASYNC_LDS
__device__ __forceinline__ void async_cp_b128(const float4* gsrc, float4* lds_dst) {
  __builtin_amdgcn_global_load_async_to_lds_b128(
      (__attribute__((address_space(1))) v4i*)(gsrc),
      (__attribute__((address_space(3))) v4i*)(lds_dst),
      /*offset=*/0, /*cpol=*/0);
}
__device__ __forceinline__ void wait_async_le8() {
#if __has_builtin(__builtin_amdgcn_s_wait_asynccnt)
  __builtin_amdgcn_s_wait_asynccnt(8);
#else
  asm volatile("s_wait_asynccnt 8" ::: "memory");
#endif
}
__device__ __forceinline__ void wait_async_le0() {
#if __has_builtin(__builtin_amdgcn_s_wait_asynccnt)
  __builtin_amdgcn_s_wait_asynccnt(0);
#else
  asm volatile("s_wait_asynccnt 0" ::: "memory");
#endif
}
#endif

// ---------------------------------------------------------------------------
// Exact dual 32-lane sum reduction via V_WMMA_F32_16X16X4_F32.
// A: lane l<16 holds A[l][0..1]=(qa_l,qb_l); lane l>=16 holds A[l-16][2..3].
// B selector: B[k][j] = 1 iff (k even) == (j<8)  =>
//   D[i][j<8]  = qa_i + qa_{i+16},  D[i][j>=8] = qb_i + qb_{i+16}.
// Sum of 8 D VGPRs + shfl_xor(16) => full 32-lane sums in cols 0..7 / 8..15.
// ---------------------------------------------------------------------------
__device__ __forceinline__ void wave_sum2(float qa, float qb, float& sa, float& sb) {
#if HAS_WMMA_F32X4
  const unsigned lane = threadIdx.x & 31u;
  const float bsel = (((lane >> 4) & 1u) == ((lane >> 3) & 1u)) ? 1.0f : 0.0f;
  v2f A; A.x = qa;   A.y = qb;
  v2f Bm; Bm.x = bsel; Bm.y = bsel;
  v8f C = {0.f, 0.f, 0.f, 0.f, 0.f, 0.f, 0.f, 0.f};
  v8f D = __builtin_amdgcn_wmma_f32_16x16x4_f32(false, A, false, Bm,
                                                (short)0, C, false, false);
  float t = D[0] + D[1] + D[2] + D[3] + D[4] + D[5] + D[6] + D[7];
  t += __shfl_xor(t, 16, 32);
  sa = __shfl(t, 0, 32);   // column 0  -> sum of qa over all 32 lanes
  sb = __shfl(t, 8, 32);   // column 8  -> sum of qb over all 32 lanes
#else
  for (int off = 16; off; off >>= 1) {
    qa += __shfl_xor(qa, off, 32);
    qb += __shfl_xor(qb, off, 32);
  }
  sa = qa; sb = qb;
#endif
}

__device__ __forceinline__ float l1_4(float4 u, float4 v) {
  return fabsf(u.x - v.x) + fabsf(u.y - v.y) + fabsf(u.z - v.z) + fabsf(u.w - v.w);
}
__device__ __forceinline__ float sq_4(float4 u, float4 v) {
  float a = u.x - v.x + EPS_PD, b = u.y - v.y + EPS_PD;
  float c = u.z - v.z + EPS_PD, d = u.w - v.w + EPS_PD;
  return a * a + b * b + c * c + d * d;
}

// ---------------------------------------------------------------------------
// Kernel 1: zero the three accumulators (d_out[1..3])
// ---------------------------------------------------------------------------
__global__ void zero_acc_kernel(float* out) {
  if (threadIdx.x < 3) out[1 + threadIdx.x] = 0.0f;
}

// ---------------------------------------------------------------------------
// Kernel 2: triplet L1 term + exemplar-center term. One wave per row.
// Each lane handles 2 float4 chunks (8 floats) of the 256-wide row.
// ---------------------------------------------------------------------------
__global__ __launch_bounds__(256) void triplet_center_kernel(
    const float* __restrict__ anchor, const float* __restrict__ positive,
    const float* __restrict__ negative, const float* __restrict__ exemplars,
    const int* __restrict__ labA, const int* __restrict__ labN,
    float* __restrict__ out) {
  const int waveId = threadIdx.x >> 5;
  const int lane   = threadIdx.x & 31;
  const int row    = blockIdx.x * 8 + waveId;   // grid sized exactly: uniform

  const float4* a = (const float4*)(anchor   + (size_t)row * DIMS);
  const float4* p = (const float4*)(positive + (size_t)row * DIMS);
  const float4* n = (const float4*)(negative + (size_t)row * DIMS);
  const int la = labA[row];
  const int ln = labN[row];
  const float4* ea = (const float4*)(exemplars + (size_t)la * DIMS);
  const float4* en = (const float4*)(exemplars + (size_t)ln * DIMS);

  float dp = 0.f, dn = 0.f;             // L1 distances
  float s_r1 = 0.f, s_n1 = 0.f;         // ||a-ea||^2, ||n-ea||^2 (with eps)
  float s_r2 = 0.f, s_n2 = 0.f;         // ||a-en||^2, ||n-en||^2 (with eps)

#pragma unroll
  for (int t = 0; t < 2; ++t) {
    const int i = lane + t * 32;        // 64 float4 per row
    float4 av = a[i], pv = p[i], nv = n[i], eav = ea[i], env = en[i];
    dp   += l1_4(av, pv);
    dn   += l1_4(av, nv);
    s_r1 += sq_4(av, eav);
    s_n1 += sq_4(nv, eav);
    s_r2 += sq_4(av, env);
    s_n2 += sq_4(nv, env);
  }

  float dpS, dnS, r1S, n1S, r2S, n2S;
  wave_sum2(dp,   dn,   dpS, dnS);
  wave_sum2(s_r1, s_n1, r1S, n1S);
  wave_sum2(s_r2, s_n2, r2S, n2S);

  if (lane == 0) {
    const float tri = fmaxf(dpS - dnS, 0.0f);
    const float cen = fmaxf(sqrtf(r1S) - sqrtf(n1S), 0.0f) +
                      fmaxf(sqrtf(n2S) - sqrtf(r2S), 0.0f);
    atomicAdd(out + 1, tri);   // losses_sum
    atomicAdd(out + 3, cen);   // loss_center
  }
}

// ---------------------------------------------------------------------------
// Kernel 3: cross-entropy (log-softmax gather, mean over 3B rows).
// One wave per row; 6 rows per wave. Row (1000 f32 = 250 float4) is staged
// into LDS with async global->LDS copies, double-buffered so row i+1 streams
// while row i is reduced (online softmax). Identity lane mapping => only
// per-wave s_wait_asynccnt, no barriers.
// ---------------------------------------------------------------------------
__global__ __launch_bounds__(256) void ce_kernel(
    const float* __restrict__ outputs, const int* __restrict__ labA,
    const int* __restrict__ labN, float* __restrict__ out) {
  const int waveId = threadIdx.x >> 5;
  const int lane   = threadIdx.x & 31;
  const int gw     = blockIdx.x * CE_WPB + waveId;
  const int row0   = gw * CE_ROWS_PER_WAVE;

#if HAS_ASYNC_LDS
  __shared__ float4 stage[CE_WPB][2][F4_PER_ROW];   // 64000 B
  {
    const float4* src = (const float4*)(outputs + (size_t)row0 * NCLS);
#pragma unroll
    for (int it = 0; it < 8; ++it) {
      const int f = lane + it * 32;
      if (f < F4_PER_ROW) async_cp_b128(src + f, &stage[waveId][0][f]);
    }
  }
#endif

  for (int i = 0; i < CE_ROWS_PER_WAVE; ++i) {
    const int row = row0 + i;

#if HAS_ASYNC_LDS
    if (i + 1 < CE_ROWS_PER_WAVE) {
      const float4* src = (const float4*)(outputs + (size_t)(row + 1) * NCLS);
#pragma unroll
      for (int it = 0; it < 8; ++it) {
        const int f = lane + it * 32;
        if (f < F4_PER_ROW) async_cp_b128(src + f, &stage[waveId][(i + 1) & 1][f]);
      }
      wait_async_le8();   // row i's 8 async ops retired (in-order completion)
    } else {
      wait_async_le0();
    }
#endif

    float m = -__builtin_huge_valf();
    float s = 0.0f;
#pragma unroll
    for (int it = 0; it < 8; ++it) {
      const int f = lane + it * 32;
      if (f < F4_PER_ROW) {
#if HAS_ASYNC_LDS
        float4 x = stage[waveId][i & 1][f];
#else
        float4 x = ((const float4*)(outputs + (size_t)row * NCLS))[f];
#endif
        const float m4 = fmaxf(fmaxf(x.x, x.y), fmaxf(x.z, x.w));
        if (m4 > m) { s *= __expf(m - m4); m = m4; }
        s += __expf(x.x - m) + __expf(x.y - m) + __expf(x.z - m) + __expf(x.w - m);
      }
    }
    // wave-level online-softmax merge
    for (int off = 16; off; off >>= 1) {
      const float mo = __shfl_xor(m, off, 32);
      const float so = __shfl_xor(s, off, 32);
      const float mn = fmaxf(m, mo);
      s = s * __expf(m - mn) + so * __expf(mo - mn);
      m = mn;
    }
    if (lane == 0) {
      const int label = (row < B_ROWS) ? labA[row]
                       : (row < 2 * B_ROWS) ? labA[row - B_ROWS]
                                            : labN[row - 2 * B_ROWS];
      const float xl = outputs[(size_t)row * NCLS + label];
      const float contrib = (m + __logf(s) - xl) * (1.0f / (float)ROWS3);
      atomicAdd(out + 2, contrib);   // loss_softmax (already mean-scaled)
    }
  }
}

// ---------------------------------------------------------------------------
// Kernel 4: compose loss_total
// ---------------------------------------------------------------------------
__global__ void finalize_kernel(float* out) {
  if (threadIdx.x == 0) {
    const float losses_sum   = out[1];
    const float loss_softmax = out[2];
    const float loss_center  = out[3];
    out[0] = loss_softmax + CENTER_W * loss_center + losses_sum;
  }
}

// ---------------------------------------------------------------------------
extern "C" void kernel_launch(void* const* d_in, const int* in_sizes, int n_in,
                              void* d_out, int out_size, void* d_ws, size_t ws_size,
                              hipStream_t stream) {
  (void)in_sizes; (void)n_in; (void)out_size; (void)d_ws; (void)ws_size;
  const float* anchor    = (const float*)d_in[0];
  const float* positive  = (const float*)d_in[1];
  const float* negative  = (const float*)d_in[2];
  const float* outputs   = (const float*)d_in[3];
  const float* exemplars = (const float*)d_in[4];
  const int*   labA      = (const int*)d_in[5];
  const int*   labN      = (const int*)d_in[6];
  float*       out       = (float*)d_out;

  hipLaunchKernelGGL(zero_acc_kernel, dim3(1), dim3(32), 0, stream, out);
  hipLaunchKernelGGL(triplet_center_kernel, dim3(B_ROWS / 8), dim3(256), 0, stream,
                     anchor, positive, negative, exemplars, labA, labN, out);
  hipLaunchKernelGGL(ce_kernel, dim3(CE_BLOCKS), dim3(256), 0, stream,
                     outputs, labA, labN, out);
  hipLaunchKernelGGL(finalize_kernel, dim3(1), dim3(32), 0, stream, out);
}